// StateVecSimTorch_9165460210293
// MI455X (gfx1250) — compile-verified
//
#include <hip/hip_runtime.h>

// ---------------------------------------------------------------------------
// 10-qubit state-vector sim, batch 32, 3 layers (complex64).
// Per layer:  S' = A5 * S * B5^T,  A5 = kron(M0..M4), B5 = kron(M5..M9),
// S = state viewed as 32x32 (row = qubits 0-4 = high bits), then ring-CNOT
// index permutation.  Complex GEMMs mapped onto V_WMMA_F32_16X16X4_F32.
// v2: planar (SoA) LDS layout + pre-negated imag planes so WMMA operands load
// directly from LDS with no de-interleave movs / inner-loop negation xors.
// One workgroup (4 waves, 128 threads) per sample; all state resident in LDS.
// ---------------------------------------------------------------------------

typedef __attribute__((ext_vector_type(2))) float v2f;
typedef __attribute__((ext_vector_type(8))) float v8f;

__device__ __forceinline__ float2 cmulf2(float2 a, float2 b) {
    return make_float2(a.x * b.x - a.y * b.y, a.x * b.y + a.y * b.x);
}

// Complex 32x32x32 GEMM: D = A * B, planar operands in LDS.
// Anim = pre-negated imag plane of A.  4 waves; wave w owns tile (w>>1, w&1).
// Uniform control flow: EXEC all-1s around every WMMA (ISA requirement).
template <bool STORE_NEG>
__device__ __forceinline__ void cgemm32x32(const float* __restrict__ Are,
                                           const float* __restrict__ Aim,
                                           const float* __restrict__ Anim,
                                           const float* __restrict__ Bre,
                                           const float* __restrict__ Bim,
                                           float* __restrict__ Dre,
                                           float* __restrict__ Dim,
                                           float* __restrict__ Dnim,
                                           int tid) {
    const int lane  = tid & 31;
    const int wv    = tid >> 5;           // 0..3, uniform per wave
    const int mo    = (wv >> 1) * 16;     // M tile offset
    const int no    = (wv & 1) * 16;      // N tile offset
    const int ln15  = lane & 15;
    const int khalf = lane >> 4;          // lane half selects K pair

    v8f cr = {};
    v8f ci = {};
#pragma unroll
    for (int kk = 0; kk < 8; ++kk) {      // K = 32 in steps of 4
        const int k = kk * 4 + 2 * khalf; // even -> 8B-aligned b64 loads
        const int m = mo + ln15;
        const int n = no + ln15;
        // A fragment (16x4 f32): VGPR0 = A[m][k], VGPR1 = A[m][k+1] — contiguous.
        v2f ar  = *(const v2f*)&Are[m * 32 + k];
        v2f ai  = *(const v2f*)&Aim[m * 32 + k];
        v2f nai = *(const v2f*)&Anim[m * 32 + k];
        // B fragment (4x16 f32): VGPR0 = B[k][n], VGPR1 = B[k+1][n] — stride-32.
        v2f br = {Bre[k * 32 + n], Bre[(k + 1) * 32 + n]};
        v2f bi = {Bim[k * 32 + n], Bim[(k + 1) * 32 + n]};
        // Cr += Ar*Br - Ai*Bi ; Ci += Ar*Bi + Ai*Br
        cr = __builtin_amdgcn_wmma_f32_16x16x4_f32(false, ar,  false, br, (short)0, cr, false, false);
        cr = __builtin_amdgcn_wmma_f32_16x16x4_f32(false, nai, false, bi, (short)0, cr, false, false);
        ci = __builtin_amdgcn_wmma_f32_16x16x4_f32(false, ar,  false, bi, (short)0, ci, false, false);
        ci = __builtin_amdgcn_wmma_f32_16x16x4_f32(false, ai,  false, br, (short)0, ci, false, false);
    }
    // D fragment (16x16 f32): VGPR r -> row mo + r + 8*khalf, col no + ln15.
#pragma unroll
    for (int r = 0; r < 8; ++r) {
        int idx = (mo + r + 8 * khalf) * 32 + no + ln15;
        Dre[idx] = cr[r];
        Dim[idx] = ci[r];
        if (STORE_NEG) Dnim[idx] = -ci[r];
    }
}

__global__ __launch_bounds__(128) void qsim10_wmma_kernel(
    const float* __restrict__ X,       // (32, 20)
    const float* __restrict__ W,       // (3,10,3) flat 90
    const float* __restrict__ Bias,    // (3,10,3) flat 90
    float* __restrict__ out)           // (32, 1024) complex64 -> interleaved f32
{
    // Planar LDS matrices (each plane 4 KB). ~44 KB total << 320 KB/WGP.
    __shared__ float sSre[1024], sSim[1024];               // state S
    __shared__ float sTre[1024], sTim[1024], sTnim[1024];  // T = A5*S (A-operand of GEMM2)
    __shared__ float sAre[1024], sAim[1024], sAnim[1024];  // A5
    __shared__ float sBre[1024], sBim[1024];               // Bop[k][n] = B5[n][k]
    __shared__ float2 sG[10][4];                           // ten 2x2 complex gates
    __shared__ int    sRing[1024];                         // ring-CNOT permutation

    const int tid = threadIdx.x;
    const int s   = blockIdx.x;
    const float* Xs = X + s * 20;

    // ---- one-time: ring permutation table + initial state |0...0> ----------
    // total[j] = perm_1(perm_2(...perm_10(j))), pairs (0,1),(1,2),...,(8,9),(9,0)
#pragma unroll
    for (int t = 0; t < 8; ++t) {
        int idx = tid + t * 128;
        int p = idx;
#pragma unroll
        for (int k = 9; k >= 0; --k) {          // apply in reverse pair order
            int c  = (k == 9) ? 9 : k;
            int tg = (k == 9) ? 0 : k + 1;
            if ((p >> (9 - c)) & 1) p ^= (1 << (9 - tg));
        }
        sRing[idx] = p;
        sSre[idx]  = (idx == 0) ? 1.0f : 0.0f;
        sSim[idx]  = 0.0f;
    }
    __syncthreads();

    for (int l = 0; l < 3; ++l) {
        // ---- 1) ten 2x2 rotation gates ------------------------------------
        if (tid < 10) {
            const int q  = tid;
            const int fb = l * 30 + q * 3;
            float phi   = Xs[(fb + 0) % 20] * W[fb + 0] + Bias[fb + 0];
            float theta = Xs[(fb + 1) % 20] * W[fb + 1] + Bias[fb + 1];
            float omega = Xs[(fb + 2) % 20] * W[fb + 2] + Bias[fb + 2];
            float st, ct;   sincosf(theta * 0.5f, &st, &ct);
            float ppo = (phi + omega) * 0.5f;
            float pmo = (phi - omega) * 0.5f;
            float sppo, cppo, spmo, cpmo;
            sincosf(ppo, &sppo, &cppo);
            sincosf(pmo, &spmo, &cpmo);
            sG[q][0] = make_float2( cppo * ct, -sppo * ct);   // e^{-i ppo} ct
            sG[q][1] = make_float2(-cpmo * st, -spmo * st);   // -e^{+i pmo} st
            sG[q][2] = make_float2( cpmo * st, -spmo * st);   // e^{-i pmo} st
            sG[q][3] = make_float2( cppo * ct,  sppo * ct);   // e^{+i ppo} ct
        }
        __syncthreads();

        // ---- 2) build A5 = kron(g0..g4), Bop = kron(g5..g9)^T (planar) -----
#pragma unroll
        for (int t = 0; t < 8; ++t) {
            int idx = tid + t * 128;
            int i = idx >> 5, j = idx & 31;
            float2 a = make_float2(1.0f, 0.0f);
            float2 b = make_float2(1.0f, 0.0f);
#pragma unroll
            for (int qq = 0; qq < 5; ++qq) {   // qubit qq is bit (4-qq) of 5-bit index
                int bi = (i >> (4 - qq)) & 1;
                int bj = (j >> (4 - qq)) & 1;
                a = cmulf2(a, sG[qq][bi * 2 + bj]);
                b = cmulf2(b, sG[5 + qq][bi * 2 + bj]);
            }
            sAre[idx]  = a.x;               // A5[i][j]
            sAim[idx]  = a.y;
            sAnim[idx] = -a.y;              // pre-negated imag plane
            sBre[j * 32 + i] = b.x;         // Bop[k][n] = B5[n][k]  (k=j, n=i)
            sBim[j * 32 + i] = b.y;
        }
        __syncthreads();

        // ---- 3) T = A5 * S ;  S' = T * B5^T  (two complex WMMA GEMMs) -----
        cgemm32x32<true >(sAre, sAim, sAnim, sSre, sSim, sTre, sTim, sTnim, tid);
        __syncthreads();
        cgemm32x32<false>(sTre, sTim, sTnim, sBre, sBim, sSre, sSim, nullptr, tid);
        __syncthreads();

        // ---- 4) ring permutation: new[j] = cur[ring[j]] (register-staged) --
        float2 v[8];
#pragma unroll
        for (int t = 0; t < 8; ++t) {
            int src = sRing[tid + t * 128];
            v[t] = make_float2(sSre[src], sSim[src]);
        }
        __syncthreads();
#pragma unroll
        for (int t = 0; t < 8; ++t) {
            int idx = tid + t * 128;
            sSre[idx] = v[t].x;
            sSim[idx] = v[t].y;
        }
        __syncthreads();
    }

    // ---- write complex64 output (interleaved re/im) ------------------------
#pragma unroll
    for (int t = 0; t < 8; ++t) {
        int idx = tid + t * 128;
        out[(s * 1024 + idx) * 2 + 0] = sSre[idx];
        out[(s * 1024 + idx) * 2 + 1] = sSim[idx];
    }
}

extern "C" void kernel_launch(void* const* d_in, const int* in_sizes, int n_in,
                              void* d_out, int out_size, void* d_ws, size_t ws_size,
                              hipStream_t stream) {
    const float* X    = (const float*)d_in[0];   // (32,20) f32
    const float* W    = (const float*)d_in[1];   // (3,10,3) f32
    const float* Bias = (const float*)d_in[2];   // (3,10,3) f32
    qsim10_wmma_kernel<<<32, 128, 0, stream>>>(X, W, Bias, (float*)d_out);
}